// SpatialTransformerAutoSize_28226525069611
// MI455X (gfx1250) — compile-verified
//
#include <hip/hip_runtime.h>
#include <stdint.h>

// ---------------------------------------------------------------------------
// 3D spatial transformer (trilinear grid_sample, zeros padding) for MI455X.
// Row-aligned mapping: 1 chunk = 1 W-row -> ix = tid (no per-point division),
// iz/iy uniform per chunk (SALU). Flow streamed via gfx1250 async
// global->LDS loads (NT hint, double-buffered per wave, ASYNCcnt-synced,
// no barriers). src gathers stay resident in the 192MB L2 (src = 39MB);
// output uses non-temporal stores so the write stream doesn't evict src.
// ---------------------------------------------------------------------------

namespace {
constexpr int Dq  = 160;
constexpr int Hq  = 192;
constexpr int Wq  = 160;
constexpr int HWq  = Hq * Wq;              // 30720
constexpr int DHWq = Dq * HWq;             // 4915200
constexpr int ROWS_PER_BATCH = Dq * Hq;    // 30720
constexpr int ROWS_TOTAL = 2 * ROWS_PER_BATCH;  // 61440

constexpr int TPB = Wq;                    // 160 threads = 5 waves, 1 row
constexpr int RPB = 16;                    // rows (chunks) per block
constexpr int NBLK = ROWS_TOTAL / RPB;     // 3840 (exact; 30720 % 16 == 0)

constexpr float SZ = 160.0f / 159.0f;      // D/(D-1)
constexpr float SY = 192.0f / 191.0f;      // H/(H-1)
constexpr float SX = 160.0f / 159.0f;      // W/(W-1)
} // namespace

__global__ __launch_bounds__(TPB) void warp3d_row_kernel(
    const float* __restrict__ src,
    const float* __restrict__ flow,
    float* __restrict__ out)
{
    // Double-buffered flow row: [buf][channel][col]. Each wave touches only
    // its own 32-lane slice -> per-wave ASYNCcnt sync, no barriers needed.
    __shared__ float fbuf[2][3][TPB];

    const int tid = threadIdx.x;

    const int rowBase = blockIdx.x * RPB;              // global row (uniform)
    const int b       = rowBase / ROWS_PER_BATCH;      // uniform per block
    const int rowInB0 = rowBase - b * ROWS_PER_BATCH;  // row within volume

    // SGPR base pointers
    const float* fZ   = flow + (size_t)b * 3 * DHWq + (size_t)rowInB0 * Wq;
    const float* fY   = fZ + DHWq;
    const float* fX   = fZ + 2 * DHWq;
    const float* srcB = src + (size_t)b * DHWq;
    float*       outB = out + (size_t)b * DHWq + (size_t)rowInB0 * Wq;

    // LDS byte addresses (low 32 bits of generic pointer == LDS offset).
    uint32_t lds[2][3];
#pragma unroll
    for (int u = 0; u < 2; ++u)
#pragma unroll
        for (int ch = 0; ch < 3; ++ch)
            lds[u][ch] = (uint32_t)(uintptr_t)&fbuf[u][ch][tid];

    const uint32_t voff0 = (uint32_t)tid * 4u;

    auto issue = [&](int cc, int bb) {
        const uint32_t voff = voff0 + (uint32_t)(cc * Wq * 4);  // 1 v_add/chunk
        asm volatile("global_load_async_to_lds_b32 %0, %1, %2 th:TH_LOAD_NT"
                     :: "v"(lds[bb][0]), "v"(voff), "s"(fZ) : "memory");
        asm volatile("global_load_async_to_lds_b32 %0, %1, %2 th:TH_LOAD_NT"
                     :: "v"(lds[bb][1]), "v"(voff), "s"(fY) : "memory");
        asm volatile("global_load_async_to_lds_b32 %0, %1, %2 th:TH_LOAD_NT"
                     :: "v"(lds[bb][2]), "v"(voff), "s"(fX) : "memory");
    };

    issue(0, 0);  // prime the pipeline

    // x-axis scalar part is invariant across chunks (ix == tid).
    const float Ax = (float)tid * SX - 0.5f;

#pragma unroll
    for (int c = 0; c < RPB; ++c) {
        const int buf = c & 1;
        if (c + 1 < RPB) {
            issue(c + 1, buf ^ 1);
            // 6 async loads in flight; the oldest 3 (chunk c) retire in order.
            asm volatile("s_wait_asynccnt 0x3" ::: "memory");
        } else {
            asm volatile("s_wait_asynccnt 0x0" ::: "memory");
        }

        const float fz = fbuf[buf][0][tid];
        const float fy = fbuf[buf][1][tid];
        const float fx = fbuf[buf][2][tid];

        // iz / iy are uniform per chunk: pure scalar math.
        const int rowInB = rowInB0 + c;
        const int iz = rowInB / Hq;
        const int iy = rowInB - iz * Hq;
        const float Az = (float)iz * SZ - 0.5f;
        const float Ay = (float)iy * SY - 0.5f;

        // coord = (idx + flow) * S - 0.5  (== reference chain to ~1 ulp)
        const float zf = fmaf(fz, SZ, Az);
        const float yf = fmaf(fy, SY, Ay);
        const float xf = fmaf(fx, SX, Ax);

        const float z0f = floorf(zf), y0f = floorf(yf), x0f = floorf(xf);
        const float wz1 = zf - z0f,  wy1 = yf - y0f,  wx1 = xf - x0f;

        const int z0 = (int)z0f, y0 = (int)y0f, x0 = (int)x0f;
        const int z1 = z0 + 1,  y1 = y0 + 1,  x1 = x0 + 1;

        // Zeros padding folded straight into the lerp weights (no mask muls).
        const float WZ1 = ((unsigned)z1 < (unsigned)Dq) ? wz1 : 0.0f;
        const float WY1 = ((unsigned)y1 < (unsigned)Hq) ? wy1 : 0.0f;
        const float WX1 = ((unsigned)x1 < (unsigned)Wq) ? wx1 : 0.0f;
        const float WZ0 = ((unsigned)z0 < (unsigned)Dq) ? (1.0f - wz1) : 0.0f;
        const float WY0 = ((unsigned)y0 < (unsigned)Hq) ? (1.0f - wy1) : 0.0f;
        const float WX0 = ((unsigned)x0 < (unsigned)Wq) ? (1.0f - wx1) : 0.0f;

        // Clamped indices -> always-legal, L2-resident loads.
        const int z0c = min(max(z0, 0), Dq - 1), z1c = min(max(z1, 0), Dq - 1);
        const int y0c = min(max(y0, 0), Hq - 1), y1c = min(max(y1, 0), Hq - 1);
        const int x0c = min(max(x0, 0), Wq - 1), x1c = min(max(x1, 0), Wq - 1);

        const int o00 = z0c * HWq + y0c * Wq;
        const int o01 = z0c * HWq + y1c * Wq;
        const int o10 = z1c * HWq + y0c * Wq;
        const int o11 = z1c * HWq + y1c * Wq;

        const float v000 = srcB[o00 + x0c];
        const float v001 = srcB[o00 + x1c];
        const float v010 = srcB[o01 + x0c];
        const float v011 = srcB[o01 + x1c];
        const float v100 = srcB[o10 + x0c];
        const float v101 = srcB[o10 + x1c];
        const float v110 = srcB[o11 + x0c];
        const float v111 = srcB[o11 + x1c];

        const float s00 = v000 * WX0 + v001 * WX1;
        const float s01 = v010 * WX0 + v011 * WX1;
        const float s10 = v100 * WX0 + v101 * WX1;
        const float s11 = v110 * WX0 + v111 * WX1;
        const float t0  = s00 * WY0 + s01 * WY1;
        const float t1  = s10 * WY0 + s11 * WY1;
        const float res = t0 * WZ0 + t1 * WZ1;

        // Non-temporal: keep src resident in L2, stream the output out.
        __builtin_nontemporal_store(res, outB + c * Wq + tid);
    }
}

extern "C" void kernel_launch(void* const* d_in, const int* in_sizes, int n_in,
                              void* d_out, int out_size, void* d_ws, size_t ws_size,
                              hipStream_t stream) {
    (void)in_sizes; (void)n_in; (void)out_size; (void)d_ws; (void)ws_size;
    const float* src  = (const float*)d_in[0];   // [2,1,160,192,160] f32
    const float* flow = (const float*)d_in[1];   // [2,3,160,192,160] f32
    float* out = (float*)d_out;                  // [2,1,160,192,160] f32

    hipLaunchKernelGGL(warp3d_row_kernel, dim3(NBLK), dim3(TPB), 0, stream,
                       src, flow, out);
}